// PointSetAbstractionLayer_24850680774733
// MI455X (gfx1250) — compile-verified
//
#include <hip/hip_runtime.h>
#include <hip/hip_bf16.h>

#define B      4
#define N      8192
#define C      64
#define COUT   128
#define NS     2048          // N / STRIDE
#define KNBR   32
#define CIN    67
#define CINP   68            // padded to multiple of 4
#define KK_STEPS 17          // CINP / 4
#define RADIUS_F 0.2f
#define R2 (RADIUS_F * RADIUS_F)
#define CPB 16               // centroids per block in stats pass

typedef __attribute__((ext_vector_type(2))) float v2f;
typedef __attribute__((ext_vector_type(8))) float v8f;

// ---------------------------------------------------------------- prep: Wp (transposed+padded W) + zero stats
__global__ void prep_kernel(const float* __restrict__ W, float* __restrict__ Wp,
                            float* __restrict__ stats) {
  int t = blockIdx.x * blockDim.x + threadIdx.x;
  if (t < 2 * COUT) stats[t] = 0.0f;
  if (t < CINP * COUT) {
    int i = t / COUT, o = t % COUT;
    Wp[t] = (i < CIN) ? W[o * CIN + i] : 0.0f;
  }
}

// ---------------------------------------------------------------- FPS: one block per batch
__global__ void __launch_bounds__(1024) fps_kernel(const float* __restrict__ points,
                                                   const long long* __restrict__ gts,
                                                   int* __restrict__ fps_idx,
                                                   float* __restrict__ cent_out,
                                                   float* __restrict__ gts_out) {
  __shared__ float rval[32];
  __shared__ int ridx[32];
  __shared__ int s_next;
  const int b = blockIdx.x;
  const int t = threadIdx.x;
  const float* P = points + (size_t)b * N * 3;

  float px[8], py[8], pz[8], dd[8];
#pragma unroll
  for (int j = 0; j < 8; j++) {
    int i = j * 1024 + t;
    px[j] = P[3 * i + 0];
    py[j] = P[3 * i + 1];
    pz[j] = P[3 * i + 2];
    dd[j] = 1e10f;
  }

  int last = 0;
  for (int s = 0; s < NS; s++) {
    if (t == 0) {
      fps_idx[b * NS + s] = last;
      cent_out[((size_t)b * NS + s) * 3 + 0] = P[3 * last + 0];
      cent_out[((size_t)b * NS + s) * 3 + 1] = P[3 * last + 1];
      cent_out[((size_t)b * NS + s) * 3 + 2] = P[3 * last + 2];
      gts_out[(size_t)b * NS + s] = (float)gts[(size_t)b * N + last];
    }
    const float lx = P[3 * last + 0];
    const float ly = P[3 * last + 1];
    const float lz = P[3 * last + 2];
    float bv = -1.0f;
    int bi = 0;
#pragma unroll
    for (int j = 0; j < 8; j++) {
      float dx = px[j] - lx, dy = py[j] - ly, dz = pz[j] - lz;
      float d = dx * dx + dy * dy + dz * dz;
      float dm = fminf(dd[j], d);
      dd[j] = dm;
      if (dm > bv) { bv = dm; bi = j * 1024 + t; }  // indices increase with j
    }
    // wave32 argmax (first-index tie-break)
#pragma unroll
    for (int off = 16; off > 0; off >>= 1) {
      float ov = __shfl_down(bv, off);
      int oi = __shfl_down(bi, off);
      if (ov > bv || (ov == bv && oi < bi)) { bv = ov; bi = oi; }
    }
    const int w = t >> 5;
    if ((t & 31) == 0) { rval[w] = bv; ridx[w] = bi; }
    __syncthreads();
    if (t == 0) {
      float fbv = rval[0];
      int fbi = ridx[0];
      for (int ww = 1; ww < 32; ww++) {
        if (rval[ww] > fbv || (rval[ww] == fbv && ridx[ww] < fbi)) { fbv = rval[ww]; fbi = ridx[ww]; }
      }
      s_next = fbi;
    }
    __syncthreads();
    last = s_next;
  }
}

// ---------------------------------------------------------------- ball query: points staged in LDS
__global__ void __launch_bounds__(256) ballq_kernel(const float* __restrict__ points,
                                                    const float* __restrict__ cent,
                                                    int* __restrict__ ballidx) {
  __shared__ float px[N], py[N], pz[N];  // 96 KB of 320 KB/WGP
  const int b = blockIdx.y;
  const int s = blockIdx.x * 256 + threadIdx.x;
  const float* P = points + (size_t)b * N * 3;
  for (int i = threadIdx.x; i < N; i += 256) {
    px[i] = P[3 * i + 0];
    py[i] = P[3 * i + 1];
    pz[i] = P[3 * i + 2];
  }
  __syncthreads();
  const size_t gs = (size_t)b * NS + s;
  const float cx = cent[gs * 3 + 0], cy = cent[gs * 3 + 1], cz = cent[gs * 3 + 2];
  int* out = ballidx + gs * KNBR;
  int cnt = 0, first = 0;
  bool havefirst = false;
  for (int i = 0; i < N; i++) {  // ascending scan == sorted-index select
    float dx = px[i] - cx, dy = py[i] - cy, dz = pz[i] - cz;
    if (dx * dx + dy * dy + dz * dz < R2) {
      if (!havefirst) { first = i; havefirst = true; }
      out[cnt++] = i;
      if (cnt == KNBR) break;
    }
  }
  for (; cnt < KNBR; cnt++) out[cnt] = first;
}

// ---------------------------------------------------------------- shared helpers for the WMMA passes
__device__ __forceinline__ void load_group(const float* __restrict__ P,
                                           const float* __restrict__ F,
                                           const int* __restrict__ nbr,
                                           float cx, float cy, float cz,
                                           float (*G)[CINP], int t) {
  const int row = t >> 3, sub = t & 7;  // 8 threads per neighbor row
  const int idx = nbr[row];
  const float* f = F + (size_t)idx * C;
  const float* p = P + (size_t)idx * 3;
  for (int c = sub; c < CINP; c += 8) {
    float v;
    if (c == 0)        v = (p[0] - cx) * (1.0f / RADIUS_F);
    else if (c == 1)   v = (p[1] - cy) * (1.0f / RADIUS_F);
    else if (c == 2)   v = (p[2] - cz) * (1.0f / RADIUS_F);
    else if (c < CIN)  v = f[c - 3];
    else               v = 0.0f;
    G[row][c] = v;
  }
}

__device__ __forceinline__ v8f wmma_tile(const float (*G)[CINP], const v2f* bf,
                                         int lane, int mt) {
  v8f acc = {};
  const int m = mt * 16 + (lane & 15);
  const int kofs = (lane >> 4) * 2;
#pragma unroll
  for (int kk = 0; kk < KK_STEPS; kk++) {
    const int kb = kk * 4 + kofs;
    v2f a;
    a.x = G[m][kb];
    a.y = G[m][kb + 1];
    acc = __builtin_amdgcn_wmma_f32_16x16x4_f32(false, a, false, bf[kk],
                                                (short)0, acc, false, false);
  }
  return acc;
}

// ---------------------------------------------------------------- pass 1: per-channel sum / sumsq of h
__global__ void __launch_bounds__(256) stats_kernel(const float* __restrict__ points,
                                                    const float* __restrict__ feats,
                                                    const float* __restrict__ cent,
                                                    const int* __restrict__ ballidx,
                                                    const float* __restrict__ Wp,
                                                    float* __restrict__ stats) {
  __shared__ float G[KNBR][CINP];
  const int t = threadIdx.x, lane = t & 31, wave = t >> 5;
  const int nt = wave;
  const int n = nt * 16 + (lane & 15);
  const int kofs = (lane >> 4) * 2;
  v2f bf[KK_STEPS];
#pragma unroll
  for (int kk = 0; kk < KK_STEPS; kk++) {
    int kb = kk * 4 + kofs;
    bf[kk].x = Wp[kb * COUT + n];
    bf[kk].y = Wp[(kb + 1) * COUT + n];
  }
  float s1 = 0.0f, s2 = 0.0f;
  for (int ci = 0; ci < CPB; ci++) {
    const int g = blockIdx.x * CPB + ci;
    const int b = g / NS;
    const float* P = points + (size_t)b * N * 3;
    const float* F = feats + (size_t)b * N * C;
    const float cx = cent[(size_t)g * 3 + 0];
    const float cy = cent[(size_t)g * 3 + 1];
    const float cz = cent[(size_t)g * 3 + 2];
    __syncthreads();
    load_group(P, F, ballidx + (size_t)g * KNBR, cx, cy, cz, G, t);
    __syncthreads();
#pragma unroll
    for (int mt = 0; mt < 2; mt++) {
      v8f acc = wmma_tile(G, bf, lane, mt);
#pragma unroll
      for (int r = 0; r < 8; r++) {
        float v = acc[r];
        s1 += v;
        s2 += v * v;
      }
    }
  }
  s1 += __shfl_xor(s1, 16);
  s2 += __shfl_xor(s2, 16);
  if (lane < 16) {
    atomicAdd(&stats[n], s1);
    atomicAdd(&stats[COUT + n], s2);
  }
}

// ---------------------------------------------------------------- fold BN into scale/shift
__global__ void finalize_kernel(const float* __restrict__ stats,
                                const float* __restrict__ gamma,
                                const float* __restrict__ beta,
                                float* __restrict__ ss) {
  const int o = threadIdx.x;
  const float cnt = (float)((size_t)B * NS * KNBR);
  const float mean = stats[o] / cnt;
  const float var = stats[COUT + o] / cnt - mean * mean;
  const float sc = gamma[o] * rsqrtf(var + 1e-5f);
  ss[o] = sc;
  ss[COUT + o] = beta[o] - mean * sc;
}

// ---------------------------------------------------------------- pass 2: recompute h, BN + ReLU + max over k
__global__ void __launch_bounds__(256) out_kernel(const float* __restrict__ points,
                                                  const float* __restrict__ feats,
                                                  const float* __restrict__ cent,
                                                  const int* __restrict__ ballidx,
                                                  const float* __restrict__ Wp,
                                                  const float* __restrict__ ss,
                                                  float* __restrict__ gf) {
  __shared__ float G[KNBR][CINP];
  const int t = threadIdx.x, lane = t & 31, wave = t >> 5;
  const int g = blockIdx.x;
  const int b = g / NS;
  const int nt = wave;
  const int n = nt * 16 + (lane & 15);
  const int kofs = (lane >> 4) * 2;
  v2f bf[KK_STEPS];
#pragma unroll
  for (int kk = 0; kk < KK_STEPS; kk++) {
    int kb = kk * 4 + kofs;
    bf[kk].x = Wp[kb * COUT + n];
    bf[kk].y = Wp[(kb + 1) * COUT + n];
  }
  const float scale = ss[n];
  const float shift = ss[COUT + n];
  const float* P = points + (size_t)b * N * 3;
  const float* F = feats + (size_t)b * N * C;
  const float cx = cent[(size_t)g * 3 + 0];
  const float cy = cent[(size_t)g * 3 + 1];
  const float cz = cent[(size_t)g * 3 + 2];
  load_group(P, F, ballidx + (size_t)g * KNBR, cx, cy, cz, G, t);
  __syncthreads();
  float mx = 0.0f;  // ReLU floor
#pragma unroll
  for (int mt = 0; mt < 2; mt++) {
    v8f acc = wmma_tile(G, bf, lane, mt);
#pragma unroll
    for (int r = 0; r < 8; r++) {
      float v = fmaxf(acc[r] * scale + shift, 0.0f);
      mx = fmaxf(mx, v);
    }
  }
  mx = fmaxf(mx, __shfl_xor(mx, 16));
  if (lane < 16) gf[(size_t)g * COUT + n] = mx;
}

// ---------------------------------------------------------------- launch
extern "C" void kernel_launch(void* const* d_in, const int* in_sizes, int n_in,
                              void* d_out, int out_size, void* d_ws, size_t ws_size,
                              hipStream_t stream) {
  (void)in_sizes; (void)n_in; (void)out_size; (void)ws_size;
  const float* points = (const float*)d_in[0];
  const float* feats = (const float*)d_in[1];
  const long long* gts = (const long long*)d_in[2];
  const float* W = (const float*)d_in[3];
  const float* gamma = (const float*)d_in[4];
  const float* beta = (const float*)d_in[5];

  float* out = (float*)d_out;
  float* cent_out = out;                          // B*NS*3
  float* gf_out = out + (size_t)B * NS * 3;       // B*NS*COUT
  float* gts_out = gf_out + (size_t)B * NS * COUT;// B*NS

  char* ws = (char*)d_ws;
  int* fps_idx = (int*)ws;              ws += sizeof(int) * (size_t)B * NS;
  int* ballidx = (int*)ws;              ws += sizeof(int) * (size_t)B * NS * KNBR;
  float* Wp = (float*)ws;               ws += sizeof(float) * CINP * COUT;
  float* stats = (float*)ws;            ws += sizeof(float) * 2 * COUT;
  float* ss = (float*)ws;

  prep_kernel<<<(CINP * COUT + 255) / 256, 256, 0, stream>>>(W, Wp, stats);
  fps_kernel<<<B, 1024, 0, stream>>>(points, gts, fps_idx, cent_out, gts_out);
  ballq_kernel<<<dim3(NS / 256, B), 256, 0, stream>>>(points, cent_out, ballidx);
  stats_kernel<<<(B * NS) / CPB, 256, 0, stream>>>(points, feats, cent_out, ballidx, Wp, stats);
  finalize_kernel<<<1, COUT, 0, stream>>>(stats, gamma, beta, ss);
  out_kernel<<<B * NS, 256, 0, stream>>>(points, feats, cent_out, ballidx, Wp, ss, gf_out);
}